// GATDrugEncoder_53326313947264
// MI455X (gfx1250) — compile-verified
//
#include <hip/hip_runtime.h>
#include <hip/hip_bf16.h>

typedef float v2f __attribute__((ext_vector_type(2)));
typedef float v8f __attribute__((ext_vector_type(8)));

#define NEG_SLOPE 0.2f
#define EPS 1e-5f

// ---------------------------------------------------------------------------
// float atomic max via sign-split int atomics (works for any sign mix,
// destination must be initialized to -inf bits 0xFF800000)
// ---------------------------------------------------------------------------
__device__ __forceinline__ void atomicMaxFloat(float* addr, float val) {
    if (val >= 0.0f) {
        atomicMax((int*)addr, __float_as_int(val));
    } else {
        atomicMin((unsigned int*)addr, __float_as_uint(val));
    }
}

// ---------------------------------------------------------------------------
// fill kernel (raw u32 pattern)
// ---------------------------------------------------------------------------
__global__ void fill_u32_kernel(unsigned int* __restrict__ p, unsigned int v, long long count) {
    long long i = (long long)blockIdx.x * blockDim.x + threadIdx.x;
    if (i < count) p[i] = v;
}

// ---------------------------------------------------------------------------
// GEMM C[M,NCOL] = A[M,K] * B[K,NCOL] using V_WMMA_F32_16X16X4_F32.
// Block = 32 * (NCOL/16) threads. grid.x = M/16 (M multiple of 16).
// Wave w computes the 16x16 tile at columns [16w, 16w+16).
// A-strip (16 x K) staged in LDS once per block.
// ---------------------------------------------------------------------------
template <int K, int NCOL>
__global__ void gemm_wmma_f32_kernel(const float* __restrict__ A,
                                     const float* __restrict__ Bm,
                                     float* __restrict__ C) {
    __shared__ float As[16 * K];
    const int rowBase = blockIdx.x * 16;
    for (int i = threadIdx.x; i < 16 * K; i += blockDim.x) {
        int r = i / K, k = i - r * K;
        As[i] = A[(long long)(rowBase + r) * K + k];
    }
    __syncthreads();

    const int wave = threadIdx.x >> 5;
    const int lane = threadIdx.x & 31;
    const int colBase = wave * 16;
    const int mn = lane & 15;           // row (for A) / col (for B, C)
    const int khalf = (lane >> 4) << 1; // 0 for lanes 0-15, 2 for lanes 16-31

    v8f acc = {};
#if __has_builtin(__builtin_amdgcn_wmma_f32_16x16x4_f32)
    for (int kk = 0; kk < K; kk += 4) {
        v2f a, b;
        // A layout (ISA 7.12.2, 32-bit A 16x4): lane<16 -> K=kk,kk+1 ; lane>=16 -> K=kk+2,kk+3
        a.x = As[mn * K + kk + khalf];
        a.y = As[mn * K + kk + khalf + 1];
        // B layout mirrored (columns of B striped across VGPRs within a lane)
        b.x = Bm[(long long)(kk + khalf) * NCOL + colBase + mn];
        b.y = Bm[(long long)(kk + khalf + 1) * NCOL + colBase + mn];
        acc = __builtin_amdgcn_wmma_f32_16x16x4_f32(false, a, false, b,
                                                    (short)0, acc, false, false);
    }
#else
    // scalar fallback (should not be taken on gfx1250)
    for (int kk = 0; kk < K; ++kk) {
        float av = As[mn * K + kk];
        for (int r = 0; r < 8; ++r)
            acc[r] += av * Bm[(long long)kk * NCOL + colBase + mn];
    }
#endif
    // C/D layout: VGPR r, lane -> row = r + 8*(lane>=16), col = lane&15
    const int mOff = (lane >> 4) << 3;
#pragma unroll
    for (int r = 0; r < 8; ++r) {
        C[(long long)(rowBase + r + mOff) * NCOL + colBase + mn] = acc[r];
    }
}

// ---------------------------------------------------------------------------
// per-(node,head) attention logits: one wave per (n,h)
// ---------------------------------------------------------------------------
__global__ void alpha_kernel(const float* __restrict__ h,
                             const float* __restrict__ a_src,
                             const float* __restrict__ a_dst,
                             float* __restrict__ as, float* __restrict__ ad,
                             int n_nodes, int heads, int C) {
    int gw = (int)(((long long)blockIdx.x * blockDim.x + threadIdx.x) >> 5);
    int lane = threadIdx.x & 31;
    if (gw >= n_nodes * heads) return;
    int hd = gw % heads;
    const float* hp = h + (long long)gw * C;
    const float* sp = a_src + hd * C;
    const float* dp = a_dst + hd * C;
    float s1 = 0.0f, s2 = 0.0f;
    for (int c = lane; c < C; c += 32) {
        float v = hp[c];
        s1 += v * sp[c];
        s2 += v * dp[c];
    }
    for (int off = 16; off; off >>= 1) {
        s1 += __shfl_down(s1, off, 32);
        s2 += __shfl_down(s2, off, 32);
    }
    if (lane == 0) { as[gw] = s1; ad[gw] = s2; }
}

// ---------------------------------------------------------------------------
// edge pass 1: e = leaky_relu(as[src]+ad[dst]); store e; atomic max per dst
// ---------------------------------------------------------------------------
__global__ void edge_max_kernel(const int* __restrict__ ei, int E, int E2, int heads,
                                const float* __restrict__ as, const float* __restrict__ ad,
                                float* __restrict__ ebuf, float* __restrict__ emax) {
    int e = (int)((long long)blockIdx.x * blockDim.x + threadIdx.x);
    if (e >= E2) return;
    int s, d;
    if (e < E) { s = ei[e]; d = ei[E + e]; } else { s = d = e - E; }
    for (int hh = 0; hh < heads; ++hh) {
        float v = as[s * heads + hh] + ad[d * heads + hh];
        v = (v > 0.0f) ? v : NEG_SLOPE * v;
        ebuf[(long long)e * heads + hh] = v;
        atomicMaxFloat(&emax[d * heads + hh], v);
    }
}

// ---------------------------------------------------------------------------
// edge pass 2: ex = exp(e - emax[dst]); store ex in place; atomic sum per dst
// ---------------------------------------------------------------------------
__global__ void edge_expsum_kernel(const int* __restrict__ ei, int E, int E2, int heads,
                                   const float* __restrict__ emax,
                                   float* __restrict__ ebuf, float* __restrict__ denom) {
    int e = (int)((long long)blockIdx.x * blockDim.x + threadIdx.x);
    if (e >= E2) return;
    int d = (e < E) ? ei[E + e] : (e - E);
    for (int hh = 0; hh < heads; ++hh) {
        float ex = __expf(ebuf[(long long)e * heads + hh] - emax[d * heads + hh]);
        ebuf[(long long)e * heads + hh] = ex;
        atomicAdd(&denom[d * heads + hh], ex);
    }
}

// ---------------------------------------------------------------------------
// edge pass 3: agg[dst] += h[src] * (ex/denom[dst]) ; one wave per edge
// ---------------------------------------------------------------------------
__global__ void edge_aggregate_kernel(const int* __restrict__ ei, int E, int E2,
                                      int heads, int F,
                                      const float* __restrict__ h,
                                      const float* __restrict__ ebuf,
                                      const float* __restrict__ denom,
                                      float* __restrict__ agg) {
    int gw = (int)(((long long)blockIdx.x * blockDim.x + threadIdx.x) >> 5);
    int lane = threadIdx.x & 31;
    if (gw >= E2) return;
    int s, d;
    if (gw < E) { s = ei[gw]; d = ei[E + gw]; } else { s = d = gw - E; }
    int C = F / heads;
    float al[2];
    for (int hh = 0; hh < heads; ++hh)
        al[hh] = ebuf[(long long)gw * heads + hh] / denom[d * heads + hh];
    const float* hs = h + (long long)s * F;
    float* ag = agg + (long long)d * F;
    for (int c = lane; c < F; c += 32) {
        int hh = c / C;
        atomicAdd(&ag[c], hs[c] * al[hh]);
    }
}

// ---------------------------------------------------------------------------
// out = relu(gamma*(x + bias - mean)*rsqrt(var+eps) + beta), in-place capable
// ---------------------------------------------------------------------------
__global__ void bias_bn_relu_kernel(const float* __restrict__ x,
                                    const float* __restrict__ b,
                                    const float* __restrict__ gamma,
                                    const float* __restrict__ beta,
                                    const float* __restrict__ mean,
                                    const float* __restrict__ var,
                                    float* __restrict__ out, long long total, int F) {
    long long i = (long long)blockIdx.x * blockDim.x + threadIdx.x;
    if (i >= total) return;
    int c = (int)(i % F);
    float v = x[i] + b[c];
    v = gamma[c] * (v - mean[c]) * rsqrtf(var[c] + EPS) + beta[c];
    out[i] = (v > 0.0f) ? v : 0.0f;
}

// ---------------------------------------------------------------------------
// global max pool: out[batch[n], c] = max over n (out pre-filled with -inf)
// ---------------------------------------------------------------------------
__global__ void pool_max_kernel(const float* __restrict__ x,
                                const int* __restrict__ batch,
                                float* __restrict__ out, int n_nodes, int F) {
    long long i = (long long)blockIdx.x * blockDim.x + threadIdx.x;
    if (i >= (long long)n_nodes * F) return;
    int n = (int)(i / F);
    int c = (int)(i - (long long)n * F);
    atomicMaxFloat(&out[batch[n] * F + c], x[i]);
}

// ---------------------------------------------------------------------------
extern "C" void kernel_launch(void* const* d_in, const int* in_sizes, int n_in,
                              void* d_out, int out_size, void* d_ws, size_t ws_size,
                              hipStream_t stream) {
    const int F_IN = 128, HID = 128, HEADS = 2;
    const int F1 = HEADS * HID; // 256
    const int N = in_sizes[0] / F_IN;   // 50000
    const int E = in_sizes[1] / 2;      // 800000
    const int E2 = E + N;               // 850000 (with self loops)

    const float* x      = (const float*)d_in[0];
    const int*   ei     = (const int*)d_in[1];
    const int*   batch  = (const int*)d_in[2];
    const float* W1     = (const float*)d_in[3];
    const float* a_src1 = (const float*)d_in[4];
    const float* a_dst1 = (const float*)d_in[5];
    const float* b1     = (const float*)d_in[6];
    const float* gamma1 = (const float*)d_in[7];
    const float* beta1  = (const float*)d_in[8];
    const float* rm1    = (const float*)d_in[9];
    const float* rv1    = (const float*)d_in[10];
    const float* W2     = (const float*)d_in[11];
    const float* a_src2 = (const float*)d_in[12];
    const float* a_dst2 = (const float*)d_in[13];
    const float* b2     = (const float*)d_in[14];
    const float* gamma2 = (const float*)d_in[15];
    const float* beta2  = (const float*)d_in[16];
    const float* rm2    = (const float*)d_in[17];
    const float* rv2    = (const float*)d_in[18];

    float* ws = (float*)d_ws;
    size_t o = 0;
    float* h1   = ws + o; o += (size_t)N * F1;   // layer1 features [N,256]
    float* aggA = ws + o; o += (size_t)N * F1;   // layer1 agg / post-BN [N,256]
    float* aggB = ws + o; o += (size_t)N * HID;  // layer2 agg / post-BN [N,128]
    float* as1  = ws + o; o += (size_t)N * HEADS;
    float* ad1  = ws + o; o += (size_t)N * HEADS;
    float* em1  = ws + o; o += (size_t)N * HEADS;
    float* dn1  = ws + o; o += (size_t)N * HEADS;
    float* as2  = ws + o; o += (size_t)N;
    float* ad2  = ws + o; o += (size_t)N;
    float* em2  = ws + o; o += (size_t)N;
    float* dn2  = ws + o; o += (size_t)N;
    float* ex1  = ws + o; o += (size_t)E2 * HEADS;
    float* ex2  = ws + o; o += (size_t)E2;
    float* h2   = h1; // h1 dead after layer-1 aggregation; reuse for [N,128]

    const unsigned int NEG_INF = 0xFF800000u;
    const unsigned int ZERO = 0u;
    auto fillBlocks = [](long long cnt) { return (unsigned)((cnt + 255) / 256); };

    // ---- init buffers -----------------------------------------------------
    fill_u32_kernel<<<fillBlocks((long long)N * HEADS), 256, 0, stream>>>((unsigned*)em1, NEG_INF, (long long)N * HEADS);
    fill_u32_kernel<<<fillBlocks((long long)N * HEADS), 256, 0, stream>>>((unsigned*)dn1, ZERO, (long long)N * HEADS);
    fill_u32_kernel<<<fillBlocks((long long)N * F1), 256, 0, stream>>>((unsigned*)aggA, ZERO, (long long)N * F1);
    fill_u32_kernel<<<fillBlocks((long long)N), 256, 0, stream>>>((unsigned*)em2, NEG_INF, (long long)N);
    fill_u32_kernel<<<fillBlocks((long long)N), 256, 0, stream>>>((unsigned*)dn2, ZERO, (long long)N);
    fill_u32_kernel<<<fillBlocks((long long)N * HID), 256, 0, stream>>>((unsigned*)aggB, ZERO, (long long)N * HID);
    fill_u32_kernel<<<fillBlocks((long long)out_size), 256, 0, stream>>>((unsigned*)d_out, NEG_INF, (long long)out_size);

    // ---- layer 1 ----------------------------------------------------------
    // h1 = x @ W1  (M=N, K=128, NCOL=256): 16 waves/block, one 16-row strip/block
    gemm_wmma_f32_kernel<128, 256><<<N / 16, 32 * (256 / 16), 0, stream>>>(x, W1, h1);
    // attention logits per (node, head)
    alpha_kernel<<<(unsigned)(((long long)N * HEADS * 32 + 255) / 256), 256, 0, stream>>>(
        h1, a_src1, a_dst1, as1, ad1, N, HEADS, HID);
    // segment softmax over incoming edges
    edge_max_kernel<<<fillBlocks(E2), 256, 0, stream>>>(ei, E, E2, HEADS, as1, ad1, ex1, em1);
    edge_expsum_kernel<<<fillBlocks(E2), 256, 0, stream>>>(ei, E, E2, HEADS, em1, ex1, dn1);
    edge_aggregate_kernel<<<(unsigned)(((long long)E2 * 32 + 255) / 256), 256, 0, stream>>>(
        ei, E, E2, HEADS, F1, h1, ex1, dn1, aggA);
    // + b1, BN, ReLU (in place)
    bias_bn_relu_kernel<<<fillBlocks((long long)N * F1), 256, 0, stream>>>(
        aggA, b1, gamma1, beta1, rm1, rv1, aggA, (long long)N * F1, F1);

    // ---- layer 2 ----------------------------------------------------------
    // h2 = aggA @ W2  (M=N, K=256, NCOL=128): 8 waves/block
    gemm_wmma_f32_kernel<256, 128><<<N / 16, 32 * (128 / 16), 0, stream>>>(aggA, W2, h2);
    alpha_kernel<<<(unsigned)(((long long)N * 32 + 255) / 256), 256, 0, stream>>>(
        h2, a_src2, a_dst2, as2, ad2, N, 1, HID);
    edge_max_kernel<<<fillBlocks(E2), 256, 0, stream>>>(ei, E, E2, 1, as2, ad2, ex2, em2);
    edge_expsum_kernel<<<fillBlocks(E2), 256, 0, stream>>>(ei, E, E2, 1, em2, ex2, dn2);
    edge_aggregate_kernel<<<(unsigned)(((long long)E2 * 32 + 255) / 256), 256, 0, stream>>>(
        ei, E, E2, 1, HID, h2, ex2, dn2, aggB);
    bias_bn_relu_kernel<<<fillBlocks((long long)N * HID), 256, 0, stream>>>(
        aggB, b2, gamma2, beta2, rm2, rv2, aggB, (long long)N * HID, HID);

    // ---- global max pool --------------------------------------------------
    pool_max_kernel<<<fillBlocks((long long)N * HID), 256, 0, stream>>>(
        aggB, batch, (float*)d_out, N, HID);
}